// PointNet2Seg_24970939859455
// MI455X (gfx1250) — compile-verified
//
#include <hip/hip_runtime.h>

typedef __attribute__((ext_vector_type(2))) float v2f;
typedef __attribute__((ext_vector_type(8))) float v8f;

#define EPS_BN 1e-5f

// ---------------------------------------------------------------------------
// Fast GEMM: Y[M,Cout] = X[M,:Cin] @ Wp[Cout,K4]^T + bias, fp32 WMMA 16x16x4.
// Requirements: M % 16 == 0, Cout % 64 == 0, K4 % 4 == 0, X rows padded with
// zeros to stride ldx (>= K4), Wp zero-padded to K4. One wave computes a
// 16x64 strip (4 N-tiles) so the A fragment is reused 4x. Branch-free K loop.
// A frag (16x4 f32): lanes 0-15 -> K=k,k+1 ; lanes 16-31 -> K=k+2,k+3
// B frag (4x16 f32): mirrored;  B[k][n] = Wp[n][k]
// C/D (16x16 f32):   vgpr r, lane<16 -> M=r ; lane>=16 -> M=r+8
// ---------------------------------------------------------------------------
__global__ void gemm_wmma_fast(const float* __restrict__ X, int ldx,
                               const float* __restrict__ Wp, int K4,
                               const float* __restrict__ bias,
                               float* __restrict__ Y, int M, int Cout) {
  int gwave  = (int)((blockIdx.x * blockDim.x + threadIdx.x) >> 5);
  int lane   = threadIdx.x & 31;
  int tilesN = Cout >> 6;                 // 64-wide strips
  int tilesM = M >> 4;
  if (gwave >= tilesM * tilesN) return;   // wave-uniform
  int tm = gwave / tilesN, tn = gwave % tilesN;
  int m0 = tm << 4, n0 = tn << 6;
  int half = lane >> 4;
  int l    = lane & 15;
  const float* Xr = X  + (size_t)(m0 + l) * ldx + 2 * half;
  const float* W0 = Wp + (size_t)(n0 + l) * K4  + 2 * half;
  const float* W1 = W0 + (size_t)16 * K4;
  const float* W2 = W0 + (size_t)32 * K4;
  const float* W3 = W0 + (size_t)48 * K4;

  v8f acc0 = {}, acc1 = {}, acc2 = {}, acc3 = {};
#pragma unroll 2
  for (int k = 0; k < K4; k += 4) {
    v2f a  = *(const v2f*)(Xr + k);
    v2f b0 = *(const v2f*)(W0 + k);
    v2f b1 = *(const v2f*)(W1 + k);
    v2f b2 = *(const v2f*)(W2 + k);
    v2f b3 = *(const v2f*)(W3 + k);
    acc0 = __builtin_amdgcn_wmma_f32_16x16x4_f32(false, a, false, b0, (short)0, acc0, false, false);
    acc1 = __builtin_amdgcn_wmma_f32_16x16x4_f32(false, a, false, b1, (short)0, acc1, false, false);
    acc2 = __builtin_amdgcn_wmma_f32_16x16x4_f32(false, a, false, b2, (short)0, acc2, false, false);
    acc3 = __builtin_amdgcn_wmma_f32_16x16x4_f32(false, a, false, b3, (short)0, acc3, false, false);
  }
  int col = n0 + l;
  float bv0 = bias[col], bv1 = bias[col + 16], bv2 = bias[col + 32], bv3 = bias[col + 48];
  float* Yr = Y + (size_t)(m0 + 8 * half) * Cout + col;
#pragma unroll
  for (int r = 0; r < 8; ++r) {
    float* yy = Yr + (size_t)r * Cout;
    yy[0]  = acc0[r] + bv0;
    yy[16] = acc1[r] + bv1;
    yy[32] = acc2[r] + bv2;
    yy[48] = acc3[r] + bv3;
  }
}

// Generic guarded WMMA GEMM (used only for head2: Cout=3).
__global__ void gemm_wmma_generic(const float* __restrict__ X, const float* __restrict__ W,
                                  const float* __restrict__ bias, float* __restrict__ Y,
                                  int M, int Cin, int Cout) {
  int gwave  = (int)((blockIdx.x * blockDim.x + threadIdx.x) >> 5);
  int lane   = threadIdx.x & 31;
  int tilesN = (Cout + 15) >> 4;
  int tilesM = (M + 15) >> 4;
  if (gwave >= tilesM * tilesN) return;
  int tm = gwave / tilesN, tn = gwave % tilesN;
  int m0 = tm << 4, n0 = tn << 4;
  int half = lane >> 4;
  int l    = lane & 15;
  int arow = m0 + l;
  int bcol = n0 + l;
  bool avalid = (arow < M);
  bool bvalid = (bcol < Cout);
  const float* Xr = X + (size_t)(avalid ? arow : 0) * Cin;
  const float* Wr = W + (size_t)(bvalid ? bcol : 0) * Cin;
  v8f acc = {};
  for (int k = 0; k < Cin; k += 4) {
    int ka = k + 2 * half;
    v2f a, b;
    a.x = (avalid && ka     < Cin) ? Xr[ka]     : 0.0f;
    a.y = (avalid && ka + 1 < Cin) ? Xr[ka + 1] : 0.0f;
    b.x = (bvalid && ka     < Cin) ? Wr[ka]     : 0.0f;
    b.y = (bvalid && ka + 1 < Cin) ? Wr[ka + 1] : 0.0f;
    acc = __builtin_amdgcn_wmma_f32_16x16x4_f32(false, a, false, b, (short)0, acc, false, false);
  }
  if (bvalid) {
    float bv = bias[bcol];
#pragma unroll
    for (int r = 0; r < 8; ++r) {
      int row = m0 + r + 8 * half;
      if (row < M) Y[(size_t)row * Cout + bcol] = acc[r] + bv;
    }
  }
}

// Zero-pad weight pack: Wp[o, k] = (k < Cin) ? W[o, k] : 0
__global__ void pack_w_kernel(const float* __restrict__ W, float* __restrict__ Wp,
                              int Cout, int Cin, int K4) {
  int t = blockIdx.x * blockDim.x + threadIdx.x;
  if (t >= Cout * K4) return;
  int o = t / K4, k = t % K4;
  Wp[t] = (k < Cin) ? W[(size_t)o * Cin + k] : 0.0f;
}

// ---------------------------------------------------------------------------
// BN batch statistics: one block per channel over all M rows.
// ---------------------------------------------------------------------------
__global__ void bn_stats_kernel(const float* __restrict__ Y, float* __restrict__ stats,
                                int M, int C) {
  int c = blockIdx.x;
  float s = 0.f, ss = 0.f;
  for (int i = threadIdx.x; i < M; i += blockDim.x) {
    float v = Y[(size_t)i * C + c];
    s += v; ss += v * v;
  }
  __shared__ float sh[512];
  sh[threadIdx.x] = s; sh[256 + threadIdx.x] = ss;
  __syncthreads();
  for (int o = 128; o > 0; o >>= 1) {
    if ((int)threadIdx.x < o) {
      sh[threadIdx.x]       += sh[threadIdx.x + o];
      sh[256 + threadIdx.x] += sh[256 + threadIdx.x + o];
    }
    __syncthreads();
  }
  if (threadIdx.x == 0) {
    float mu  = sh[0] / (float)M;
    float var = sh[256] / (float)M - mu * mu;
    stats[2 * c]     = mu;
    stats[2 * c + 1] = rsqrtf(fmaxf(var, 0.f) + EPS_BN);
  }
}

__global__ void bn_apply_kernel(float* __restrict__ Y, const float* __restrict__ stats,
                                const float* __restrict__ g, const float* __restrict__ be,
                                int total, int C) {
  int t = blockIdx.x * blockDim.x + threadIdx.x;
  if (t >= total) return;
  int c = t % C;
  float v = (Y[t] - stats[2 * c]) * stats[2 * c + 1] * g[c] + be[c];
  Y[t] = fmaxf(v, 0.f);
}

// ---------------------------------------------------------------------------
// Farthest point sampling: one workgroup per batch; running min-dist in LDS.
// ---------------------------------------------------------------------------
__global__ void fps_kernel(const float* __restrict__ xyz, int* __restrict__ cents,
                           int N, int npoint) {
  int b = blockIdx.x;
  const float* P = xyz + (size_t)b * N * 3;
  __shared__ float sdist[4096];
  __shared__ float rmax[256];
  __shared__ int   rarg[256];
  __shared__ int   sfar;
  for (int i = threadIdx.x; i < N; i += blockDim.x) sdist[i] = 1e10f;
  if (threadIdx.x == 0) sfar = 0;
  __syncthreads();
  for (int it = 0; it < npoint; ++it) {
    int far = sfar;
    if (threadIdx.x == 0) cents[b * npoint + it] = far;
    float cx = P[far * 3 + 0], cy = P[far * 3 + 1], cz = P[far * 3 + 2];
    float best = -1.0f; int bi = 0;
    for (int i = threadIdx.x; i < N; i += blockDim.x) {
      float dx = P[i * 3 + 0] - cx, dy = P[i * 3 + 1] - cy, dz = P[i * 3 + 2] - cz;
      float nd = fminf(sdist[i], dx * dx + dy * dy + dz * dz);
      sdist[i] = nd;
      if (nd > best) { best = nd; bi = i; }
    }
    rmax[threadIdx.x] = best; rarg[threadIdx.x] = bi;
    __syncthreads();
    for (int o = 128; o > 0; o >>= 1) {
      if ((int)threadIdx.x < o && rmax[threadIdx.x + o] > rmax[threadIdx.x]) {
        rmax[threadIdx.x] = rmax[threadIdx.x + o];
        rarg[threadIdx.x] = rarg[threadIdx.x + o];
      }
      __syncthreads();
    }
    if (threadIdx.x == 0) sfar = rarg[0];
    __syncthreads();
  }
}

__global__ void gather_rows_kernel(const float* __restrict__ src, const int* __restrict__ idx,
                                   float* __restrict__ dst, int Nsrc, int Sper, int C, int total) {
  int t = blockIdx.x * blockDim.x + threadIdx.x;
  if (t >= total) return;
  int b = t / Sper;
  int j = idx[t];
  const float* s = src + ((size_t)b * Nsrc + j) * C;
  float* d = dst + (size_t)t * C;
  for (int c = 0; c < C; ++c) d[c] = s[c];
}

// First <=K indices (ascending) within radius, padded with first hit.
__global__ void ball_query_kernel(const float* __restrict__ xyz, const float* __restrict__ cen,
                                  int* __restrict__ idx, int Nsrc, int S, int K, float r2, int total) {
  int t = blockIdx.x * blockDim.x + threadIdx.x;
  if (t >= total) return;
  int b = t / S;
  const float* P = xyz + (size_t)b * Nsrc * 3;
  float cx = cen[t * 3], cy = cen[t * 3 + 1], cz = cen[t * 3 + 2];
  int* o = idx + (size_t)t * K;
  int cnt = 0;
  for (int j = 0; j < Nsrc && cnt < K; ++j) {
    float dx = P[j * 3] - cx, dy = P[j * 3 + 1] - cy, dz = P[j * 3 + 2] - cz;
    if (dx * dx + dy * dy + dz * dz <= r2) o[cnt++] = j;
  }
  int first = (cnt > 0) ? o[0] : 0;
  for (int q = cnt; q < K; ++q) o[q] = first;
}

// feat row (stride Cd, zero-padded) = [xyz[idx]-center, points[idx], 0...]
__global__ void group_feat_kernel(const float* __restrict__ xyz, const float* __restrict__ cen,
                                  const float* __restrict__ pts, const int* __restrict__ idx,
                                  float* __restrict__ feat, int Nsrc, int S, int K, int Cp,
                                  int Cd, int total) {
  int t = blockIdx.x * blockDim.x + threadIdx.x;
  if (t >= total) return;
  int bs = t / K;            // b*S + s
  int b  = bs / S;
  int j  = idx[t];
  const float* p = xyz + ((size_t)b * Nsrc + j) * 3;
  const float* c = cen + (size_t)bs * 3;
  float* o = feat + (size_t)t * Cd;
  o[0] = p[0] - c[0]; o[1] = p[1] - c[1]; o[2] = p[2] - c[2];
  for (int q = 0; q < Cp; ++q) o[3 + q] = pts[((size_t)b * Nsrc + j) * Cp + q];
  for (int q = 3 + Cp; q < Cd; ++q) o[q] = 0.0f;
}

// y[p,c] = max_s x[p,s,c]
__global__ void maxpool_kernel(const float* __restrict__ x, float* __restrict__ y,
                               int P, int S, int C) {
  int t = blockIdx.x * blockDim.x + threadIdx.x;
  if (t >= P * C) return;
  int p = t / C, c = t % C;
  const float* b = x + ((size_t)p * S) * C + c;
  float m = b[0];
  for (int s = 1; s < S; ++s) m = fmaxf(m, b[(size_t)s * C]);
  y[(size_t)p * C + c] = m;
}

__global__ void copy_cols_kernel(const float* __restrict__ src, float* __restrict__ dst,
                                 int M, int Cs, int Cd, int off) {
  int t = blockIdx.x * blockDim.x + threadIdx.x;
  if (t >= M * Cs) return;
  int m = t / Cs, c = t % Cs;
  dst[(size_t)m * Cd + off + c] = src[t];
}

__global__ void zero_cols_kernel(float* __restrict__ dst, int M, int Cd, int start, int cnt) {
  int t = blockIdx.x * blockDim.x + threadIdx.x;
  if (t >= M * cnt) return;
  int m = t / cnt, c = t % cnt;
  dst[(size_t)m * Cd + start + c] = 0.0f;
}

// dst[m, off+c] = src[b, c] with b = m / rows_per_b  (broadcast of l3p)
__global__ void bcast_rows_kernel(const float* __restrict__ src, float* __restrict__ dst,
                                  int M, int rows_per_b, int Cs, int Cd, int off) {
  int t = blockIdx.x * blockDim.x + threadIdx.x;
  if (t >= M * Cs) return;
  int m = t / Cs, c = t % Cs;
  int b = m / rows_per_b;
  dst[(size_t)m * Cd + off + c] = src[(size_t)b * Cs + c];
}

// 3-NN (sq dist) + normalized inverse-distance weights
__global__ void knn3_kernel(const float* __restrict__ q, const float* __restrict__ r,
                            int* __restrict__ oidx, float* __restrict__ ow,
                            int N1, int N2, int total) {
  int t = blockIdx.x * blockDim.x + threadIdx.x;
  if (t >= total) return;
  int b = t / N1;
  float qx = q[t * 3], qy = q[t * 3 + 1], qz = q[t * 3 + 2];
  const float* R = r + (size_t)b * N2 * 3;
  float d0 = 1e30f, d1 = 1e30f, d2 = 1e30f; int i0 = 0, i1 = 0, i2 = 0;
  for (int j = 0; j < N2; ++j) {
    float dx = R[j * 3] - qx, dy = R[j * 3 + 1] - qy, dz = R[j * 3 + 2] - qz;
    float d = dx * dx + dy * dy + dz * dz;
    if (d < d0)      { d2 = d1; i2 = i1; d1 = d0; i1 = i0; d0 = d; i0 = j; }
    else if (d < d1) { d2 = d1; i2 = i1; d1 = d;  i1 = j; }
    else if (d < d2) { d2 = d;  i2 = j; }
  }
  float w0 = 1.f / (d0 + 1e-8f), w1 = 1.f / (d1 + 1e-8f), w2 = 1.f / (d2 + 1e-8f);
  float s = w0 + w1 + w2;
  oidx[t * 3] = i0; oidx[t * 3 + 1] = i1; oidx[t * 3 + 2] = i2;
  ow[t * 3] = w0 / s; ow[t * 3 + 1] = w1 / s; ow[t * 3 + 2] = w2 / s;
}

// dst[m, off+c] = sum_j w_j * pts2[b, idx_j, c]
__global__ void interp_kernel(const float* __restrict__ pts2, const int* __restrict__ kidx,
                              const float* __restrict__ kw, float* __restrict__ dst,
                              int N1, int N2, int C, int Cd, int off, int total) {
  int t = blockIdx.x * blockDim.x + threadIdx.x;
  if (t >= total) return;
  int c = t % C;
  int m = t / C;
  int b = m / N1;
  const float* P2 = pts2 + (size_t)b * N2 * C;
  int i0 = kidx[m * 3], i1 = kidx[m * 3 + 1], i2 = kidx[m * 3 + 2];
  float v = kw[m * 3]     * P2[(size_t)i0 * C + c]
          + kw[m * 3 + 1] * P2[(size_t)i1 * C + c]
          + kw[m * 3 + 2] * P2[(size_t)i2 * C + c];
  dst[(size_t)m * Cd + off + c] = v;
}

// (B,N,C) -> (B,C,N)
__global__ void transpose_out_kernel(const float* __restrict__ y, float* __restrict__ out,
                                     int B, int N, int C) {
  int t = blockIdx.x * blockDim.x + threadIdx.x;
  if (t >= B * N * C) return;
  int c = t % C;
  int n = (t / C) % N;
  int b = t / (C * N);
  out[((size_t)b * C + c) * N + n] = y[t];
}

// ---------------------------------------------------------------------------
extern "C" void kernel_launch(void* const* d_in, const int* in_sizes, int n_in,
                              void* d_out, int out_size, void* d_ws, size_t ws_size,
                              hipStream_t stream) {
  (void)in_sizes; (void)n_in; (void)out_size; (void)ws_size;
  const int B = 16, N = 4096;
  const float* xyz = (const float*)d_in[0];
  auto W = [&](int i) { return (const float*)d_in[i]; };

  // ---- workspace carve (floats), ~290 MB total ----
  float* base = (float*)d_ws;
  size_t off = 0;
  auto allocf = [&](size_t n) { float* p = base + off; off += (n + 63) & ~(size_t)63; return p; };
  float* bufA  = allocf(34000000);               // ping
  float* bufB  = allocf(34000000);               // pong
  float* wpack = allocf(524288);                 // packed weights (max 1024x512)
  float* l1x   = allocf((size_t)B * 512 * 3);
  float* l2x   = allocf((size_t)B * 128 * 3);
  float* l1p   = allocf((size_t)B * 512 * 128);
  float* l2p   = allocf((size_t)B * 128 * 256);
  float* l2p2  = allocf((size_t)B * 128 * 256);  // fp3 output
  float* l1p2  = allocf((size_t)B * 512 * 128);  // fp2 output
  float* l3p   = allocf((size_t)B * 1024);
  float* stats = allocf(2 * 1024);
  int* cents1 = (int*)allocf((size_t)B * 512);
  int* cents2 = (int*)allocf((size_t)B * 128);
  int* idx1   = (int*)allocf((size_t)B * 512 * 32);
  int* idx2   = (int*)allocf((size_t)B * 128 * 64);
  int* kidx   = (int*)allocf((size_t)B * 4096 * 3);
  float* kw   = allocf((size_t)B * 4096 * 3);

  auto cdiv = [](long a, long b) { return (int)((a + b - 1) / b); };
  auto bn = [&](float* Y, int wi, int M, int C) {
    bn_stats_kernel<<<C, 256, 0, stream>>>(Y, stats, M, C);
    bn_apply_kernel<<<cdiv((long)M * C, 256), 256, 0, stream>>>(Y, stats, W(wi + 2), W(wi + 3), M * C, C);
  };
  // padded-stride MLP layer: X has row stride ldx (zero padded to >= K4)
  auto mlp = [&](const float* X, int ldx, float* Y, int M, int Cin, int Cout, int wi) {
    int K4 = (Cin + 3) & ~3;
    pack_w_kernel<<<cdiv((long)Cout * K4, 256), 256, 0, stream>>>(W(wi), wpack, Cout, Cin, K4);
    long waves = (long)(M / 16) * (Cout / 64);
    gemm_wmma_fast<<<cdiv(waves * 32, 256), 256, 0, stream>>>(X, ldx, wpack, K4, W(wi + 1), Y, M, Cout);
    bn(Y, wi, M, Cout);
  };

  // ---- SA1: 4096 -> 512 points, r=0.2, K=32, MLP [64,64,128] ----
  fps_kernel<<<B, 256, 0, stream>>>(xyz, cents1, N, 512);
  gather_rows_kernel<<<cdiv(B * 512, 256), 256, 0, stream>>>(xyz, cents1, l1x, N, 512, 3, B * 512);
  ball_query_kernel<<<cdiv(B * 512, 256), 256, 0, stream>>>(xyz, l1x, idx1, N, 512, 32, 0.04f, B * 512);
  group_feat_kernel<<<cdiv(B * 512 * 32, 256), 256, 0, stream>>>(xyz, l1x, (const float*)nullptr, idx1,
                                                                 bufA, N, 512, 32, 0, 4, B * 512 * 32);
  const int M1 = B * 512 * 32;
  mlp(bufA, 4, bufB, M1, 3, 64, 1);
  mlp(bufB, 64, bufA, M1, 64, 64, 5);
  mlp(bufA, 64, bufB, M1, 64, 128, 9);
  maxpool_kernel<<<cdiv((long)B * 512 * 128, 256), 256, 0, stream>>>(bufB, l1p, B * 512, 32, 128);

  // ---- SA2: 512 -> 128 points, r=0.4, K=64, MLP [128,128,256] ----
  fps_kernel<<<B, 256, 0, stream>>>(l1x, cents2, 512, 128);
  gather_rows_kernel<<<cdiv(B * 128, 256), 256, 0, stream>>>(l1x, cents2, l2x, 512, 128, 3, B * 128);
  ball_query_kernel<<<cdiv(B * 128, 256), 256, 0, stream>>>(l1x, l2x, idx2, 512, 128, 64, 0.16f, B * 128);
  group_feat_kernel<<<cdiv(B * 128 * 64, 256), 256, 0, stream>>>(l1x, l2x, l1p, idx2,
                                                                 bufA, 512, 128, 64, 128, 132, B * 128 * 64);
  const int M2 = B * 128 * 64;
  mlp(bufA, 132, bufB, M2, 131, 128, 13);
  mlp(bufB, 128, bufA, M2, 128, 128, 17);
  mlp(bufA, 128, bufB, M2, 128, 256, 21);
  maxpool_kernel<<<cdiv((long)B * 128 * 256, 256), 256, 0, stream>>>(bufB, l2p, B * 128, 64, 256);

  // ---- SA3 (global): concat [l2x,l2p] (stride 260, pad col) -> MLP [256,512,1024] ----
  const int M3 = B * 128;
  copy_cols_kernel<<<cdiv((long)M3 * 3, 256), 256, 0, stream>>>(l2x, bufA, M3, 3, 260, 0);
  copy_cols_kernel<<<cdiv((long)M3 * 256, 256), 256, 0, stream>>>(l2p, bufA, M3, 256, 260, 3);
  zero_cols_kernel<<<cdiv((long)M3, 256), 256, 0, stream>>>(bufA, M3, 260, 259, 1);
  mlp(bufA, 260, bufB, M3, 259, 256, 25);
  mlp(bufB, 256, bufA, M3, 256, 512, 29);
  mlp(bufA, 512, bufB, M3, 512, 1024, 33);
  maxpool_kernel<<<cdiv((long)B * 1024, 256), 256, 0, stream>>>(bufB, l3p, B, 128, 1024);

  // ---- FP3: concat [l2p, broadcast(l3p)] (stride 1280) -> MLP [256,256] ----
  copy_cols_kernel<<<cdiv((long)M3 * 256, 256), 256, 0, stream>>>(l2p, bufA, M3, 256, 1280, 0);
  bcast_rows_kernel<<<cdiv((long)M3 * 1024, 256), 256, 0, stream>>>(l3p, bufA, M3, 128, 1024, 1280, 256);
  mlp(bufA, 1280, bufB, M3, 1280, 256, 37);
  mlp(bufB, 256, l2p2, M3, 256, 256, 41);

  // ---- FP2: interp(512<-128) + concat l1p (stride 384) -> MLP [256,128] ----
  knn3_kernel<<<cdiv(B * 512, 256), 256, 0, stream>>>(l1x, l2x, kidx, kw, 512, 128, B * 512);
  const int M4 = B * 512;
  copy_cols_kernel<<<cdiv((long)M4 * 128, 256), 256, 0, stream>>>(l1p, bufA, M4, 128, 384, 0);
  interp_kernel<<<cdiv((long)M4 * 256, 256), 256, 0, stream>>>(l2p2, kidx, kw, bufA, 512, 128, 256, 384, 128, M4 * 256);
  mlp(bufA, 384, bufB, M4, 384, 256, 45);
  mlp(bufB, 256, l1p2, M4, 256, 128, 49);

  // ---- FP1: interp(4096<-512) (stride 128) -> MLP [128,128] ----
  knn3_kernel<<<cdiv(B * 4096, 256), 256, 0, stream>>>(xyz, l1x, kidx, kw, 4096, 512, B * 4096);
  const int M5 = B * 4096;
  interp_kernel<<<cdiv((long)M5 * 128, 256), 256, 0, stream>>>(l1p2, kidx, kw, bufA, 4096, 512, 128, 128, 0, M5 * 128);
  mlp(bufA, 128, bufB, M5, 128, 128, 53);
  mlp(bufB, 128, bufA, M5, 128, 128, 57);

  // ---- heads ----
  mlp(bufA, 128, bufB, M5, 128, 128, 61);                 // head1 + BN + ReLU
  {
    long waves = (long)((M5 + 15) / 16) * ((3 + 15) / 16);
    gemm_wmma_generic<<<cdiv(waves * 32, 256), 256, 0, stream>>>(bufB, W(65), W(66), bufA, M5, 128, 3);
  }
  transpose_out_kernel<<<cdiv((long)M5 * 3, 256), 256, 0, stream>>>(bufA, (float*)d_out, B, 4096, 3);
}